// DynamicKernel_8409545965635
// MI455X (gfx1250) — compile-verified
//
#include <hip/hip_runtime.h>

// Fused adaptive-conv kernel for MI455X (gfx1250, wave32).
// out = Σ_taps x_nbr · tanh(conv3x3_9ch(x) + b), fully fused: one pass over x.
// Conv stage runs on V_WMMA_F32_16X16X4_F32 (chain of 3, K=12 zero-padded to cover 9 taps).
// D[ch][pix] orientation => lane n holds pixel n's channels 0..7 locally; ch8 via one shfl_xor.

typedef float v2f __attribute__((ext_vector_type(2)));
typedef float v8f __attribute__((ext_vector_type(8)));

#define IMG  512
#define TX   128   // tile width  (pixels per wave-row)
#define TY   8     // tile height (one wave per row, 8 waves = 256 threads)
#define LDSW 132   // padded LDS row stride (row length is TX+2=130)

__device__ __forceinline__ float fast_tanh(float x) {
#if __has_builtin(__builtin_amdgcn_tanhf)
  return __builtin_amdgcn_tanhf(x);            // native v_tanh_f32 on CDNA5
#else
  float e = __builtin_amdgcn_exp2f(x * 2.885390081777927f);  // exp(2x)
  return 1.0f - 2.0f / (e + 1.0f);
#endif
}

__global__ __launch_bounds__(256) void adaconv_fused(
    const float* __restrict__ x, const float* __restrict__ W,
    const float* __restrict__ b, float* __restrict__ out) {
  __shared__ float sm[(TY + 2) * LDSW];

  const int tid = threadIdx.x;
  const int bx  = blockIdx.x * TX;
  const int by  = blockIdx.y * TY;
  const float* xb = x + (size_t)blockIdx.z * IMG * IMG;

  // Cooperative load of (TY+2) x (TX+2) input tile with zero halo (pad=1).
  // sm[r][c] = x[by + r - 1, bx + c - 1] (0 outside image).
  for (int t = tid; t < (TY + 2) * (TX + 2); t += 256) {
    int r  = t / (TX + 2);
    int c  = t - r * (TX + 2);
    int gy = by + r - 1;
    int gx = bx + c - 1;
    float v = 0.0f;
    if ((unsigned)gy < IMG && (unsigned)gx < IMG) v = xb[gy * IMG + gx];
    sm[r * LDSW + c] = v;
  }

  const int lane = tid & 31;
  const int n    = lane & 15;   // pixel-in-group / weight row (channel)
  const int half = lane >> 4;   // K sub-slice selector for f32 WMMA operand layout
  const int wv   = tid >> 5;    // wave id == output row inside tile

  // A operand (wave-invariant): A[ch][k] = W[ch*9 + k], zero-padded to 16x12.
  // f32 A 16x4 layout: lanes 0-15 rows M=0..15 hold K={0,1}; lanes 16-31 hold K={2,3}.
  v2f a[3];
#pragma unroll
  for (int q = 0; q < 3; ++q) {
    int k0 = 4 * q + 2 * half;
    a[q].x = (n < 9 && k0     < 9) ? W[n * 9 + k0    ] : 0.0f;
    a[q].y = (n < 9 && k0 + 1 < 9) ? W[n * 9 + k0 + 1] : 0.0f;
  }
  // C init = bias broadcast along pixels: lane (n,half) VGPR r = b[r + 8*half].
  v8f cb;
#pragma unroll
  for (int r = 0; r < 8; ++r) {
    int ch = r + 8 * half;
    cb[r] = (ch < 9) ? b[ch] : 0.0f;
  }

  __syncthreads();

  const int gy = by + wv;

#pragma unroll
  for (int g = 0; g < TX / 16; ++g) {
    const int col = g * 16 + n;  // sm column of tap dx=0 for this pixel

    // B operand: Nbr[k][pix] = sm[wv + k/3][col + k%3] (taps k=0..8, else 0).
    v2f bq[3];
#pragma unroll
    for (int q = 0; q < 3; ++q) {
      int k0 = 4 * q + 2 * half;
      int t0 = (k0     < 9) ? k0     : 8;   // clamp so LDS addr stays in-tile
      int t1 = (k0 + 1 < 9) ? k0 + 1 : 8;
      float v0 = sm[(wv + t0 / 3) * LDSW + col + t0 % 3];
      float v1 = sm[(wv + t1 / 3) * LDSW + col + t1 % 3];
      bq[q].x = (k0     < 9) ? v0 : 0.0f;
      bq[q].y = (k0 + 1 < 9) ? v1 : 0.0f;
    }

    // D = W x Nbr + bias via 3 chained fp32 WMMAs (K = 4 each).
    v8f d;
    d = __builtin_amdgcn_wmma_f32_16x16x4_f32(false, a[0], false, bq[0], (short)0, cb, false, false);
    d = __builtin_amdgcn_wmma_f32_16x16x4_f32(false, a[1], false, bq[1], (short)0, d,  false, false);
    d = __builtin_amdgcn_wmma_f32_16x16x4_f32(false, a[2], false, bq[2], (short)0, d,  false, false);

    // Channel 8 of pixel n sits in lane n+16's d[0]; fetch across halves.
    float t8 = fast_tanh(__shfl_xor(d[0], 16, 32));

    // out[pix] = Σ_c tanh(D[c][pix]) * nbr(perm(c)),  perm(3i+j) = 3j+i (tap transpose).
    float acc = t8 * sm[(wv + 2) * LDSW + col + 2];  // c=8 -> tap (2,2)
    const int PERM[8] = {0, 3, 6, 1, 4, 7, 2, 5};
#pragma unroll
    for (int c = 0; c < 8; ++c) {
      int tp = PERM[c];
      acc += fast_tanh(d[c]) * sm[(wv + tp / 3) * LDSW + col + tp % 3];
    }

    if (half == 0) {
      int gx = bx + g * 16 + n;
      out[((size_t)blockIdx.z * IMG + gy) * IMG + gx] = acc;   // coalesced 16-wide
    }
  }
}

extern "C" void kernel_launch(void* const* d_in, const int* in_sizes, int n_in,
                              void* d_out, int out_size, void* d_ws, size_t ws_size,
                              hipStream_t stream) {
  const float* x = (const float*)d_in[0];   // (B,1,512,512) fp32
  const float* W = (const float*)d_in[1];   // (9,1,3,3)     fp32
  const float* b = (const float*)d_in[2];   // (9,)          fp32
  float* out = (float*)d_out;               // (B,1,512,512) fp32

  const int batch = in_sizes[0] / (IMG * IMG);
  dim3 grid(IMG / TX, IMG / TY, batch);
  adaconv_fused<<<grid, dim3(256), 0, stream>>>(x, W, b, out);
}